// PAVREncoderDecoder_13709535608832
// MI455X (gfx1250) — compile-verified
//
#include <hip/hip_runtime.h>
#include <hip/hip_bf16.h>
#include <math.h>

#define N_NODES 512
#define HCDIM   256
#define NHEAD   8
#define CCH     32
#define NEDGE   16384
#define ETOT    (NEDGE + N_NODES)
#define IN_LEN  17929
#define OFF_DIST   16384
#define OFF_MARKOV 16896
#define OFF_WEEK   17408
#define OFF_CAP    17411
#define OFF_VEH    17414
#define OFF_STOP   17417

typedef float v2f __attribute__((ext_vector_type(2)));
typedef float v8f __attribute__((ext_vector_type(8)));

// ---------------------------------------------------------------- utilities
__global__ void k_fill(float* __restrict__ p, int n, float v) {
    int i = blockIdx.x * blockDim.x + threadIdx.x;
    if (i < n) p[i] = v;
}

__global__ void k_relu(float* __restrict__ p, int n) {
    int i = blockIdx.x * blockDim.x + threadIdx.x;
    if (i < n) p[i] = fmaxf(p[i], 0.0f);
}

__device__ __forceinline__ void atomicMaxF(float* addr, float val) {
    // valid total order for IEEE floats: ints for >=0, unsigned-min for <0
    if (val >= 0.0f) atomicMax((int*)addr, __float_as_int(val));
    else             atomicMin((unsigned int*)addr, __float_as_uint(val));
}

// ---------------------------------------------------------------- dist min/max
__global__ void k_init_minmax(unsigned* mnmx) {
    mnmx[0] = 0x7f800000u;  // +inf
    mnmx[1] = 0u;           // 0.0f (dist is uniform[0,1): non-negative)
}

__global__ void k_minmax(const float* __restrict__ x, int n, unsigned* mnmx) {
    __shared__ unsigned smn[256], smx[256];
    int t = threadIdx.x;
    unsigned mn = 0x7f800000u, mx = 0u;
    for (int i = blockIdx.x * blockDim.x + t; i < n; i += gridDim.x * blockDim.x) {
        unsigned b = __float_as_uint(x[i]);   // non-negative floats order as uints
        mn = min(mn, b); mx = max(mx, b);
    }
    smn[t] = mn; smx[t] = mx; __syncthreads();
    for (int s = 128; s > 0; s >>= 1) {
        if (t < s) { smn[t] = min(smn[t], smn[t + s]); smx[t] = max(smx[t], smx[t + s]); }
        __syncthreads();
    }
    if (t == 0) { atomicMin(&mnmx[0], smn[0]); atomicMax(&mnmx[1], smx[0]); }
}

__global__ void k_dist_norm(const float* __restrict__ dist, const unsigned* __restrict__ mnmx,
                            float* __restrict__ dn) {
    int i = blockIdx.x * blockDim.x + threadIdx.x;
    float mn = __uint_as_float(mnmx[0]);
    float mx = __uint_as_float(mnmx[1]);
    float inv = 1.0f / (mx - mn);
    dn[i] = (dist[i] - mn) * inv;
}

// ---------------------------------------------------------------- edge prep
__global__ void k_edge_prep(const int* __restrict__ ei, const float* __restrict__ markov,
                            float* __restrict__ ea, float* __restrict__ deg,
                            float* __restrict__ loopsum) {
    int e = blockIdx.x * blockDim.x + threadIdx.x;
    if (e >= NEDGE) return;
    int r = ei[e], c = ei[NEDGE + e];
    float a = markov[(size_t)r * N_NODES + c];
    ea[e] = a;
    atomicAdd(&deg[c], 1.0f);
    atomicAdd(&loopsum[c], a);
}

__global__ void k_loop_attr(const float* __restrict__ loopsum, const float* __restrict__ deg,
                            float* __restrict__ la) {
    int i = blockIdx.x * blockDim.x + threadIdx.x;
    if (i < N_NODES) la[i] = loopsum[i] / fmaxf(deg[i], 1.0f);
}

// ---------------------------------------------------------------- GAT pieces
__global__ void k_h_layer1(const float* __restrict__ demand, const float* __restrict__ W,
                           float* __restrict__ h) {
    int i = blockIdx.x, t = threadIdx.x;
    h[i * HCDIM + t] = demand[i] * W[t];   // x is N x 1, W is (HC,1)
}

__global__ void k_attn_scores(const float* __restrict__ h, const float* __restrict__ att_src,
                              const float* __restrict__ att_dst,
                              float* __restrict__ a_src, float* __restrict__ a_dst) {
    int idx = blockIdx.x * blockDim.x + threadIdx.x;   // (node, head)
    if (idx >= N_NODES * NHEAD) return;
    int i = idx >> 3, hh = idx & 7;
    const float* hr = h + (size_t)i * HCDIM + hh * CCH;
    float s = 0.0f, d = 0.0f;
    for (int c = 0; c < CCH; ++c) {
        float v = hr[c];
        s += v * att_src[hh * CCH + c];
        d += v * att_dst[hh * CCH + c];
    }
    a_src[idx] = s; a_dst[idx] = d;
}

__global__ void k_wedot(const float* __restrict__ We, const float* __restrict__ att_edge,
                        float* __restrict__ wd) {
    int hh = threadIdx.x;
    if (hh >= NHEAD) return;
    float s = 0.0f;
    for (int c = 0; c < CCH; ++c) s += We[hh * CCH + c] * att_edge[hh * CCH + c];
    wd[hh] = s;
}

__global__ void k_edge_alpha(const int* __restrict__ ei, const float* __restrict__ ea,
                             const float* __restrict__ la, const float* __restrict__ a_src,
                             const float* __restrict__ a_dst, const float* __restrict__ wd,
                             float* __restrict__ alpha, float* __restrict__ amax) {
    int idx = blockIdx.x * blockDim.x + threadIdx.x;
    if (idx >= ETOT * NHEAD) return;
    int e = idx >> 3, hh = idx & 7;
    int r, c; float av;
    if (e < NEDGE) { r = ei[e]; c = ei[NEDGE + e]; av = ea[e]; }
    else           { r = c = e - NEDGE; av = la[e - NEDGE]; }
    float al = a_src[r * NHEAD + hh] + a_dst[c * NHEAD + hh] + av * wd[hh];
    al = (al > 0.0f) ? al : 0.2f * al;          // leaky_relu(., 0.2)
    alpha[idx] = al;
    atomicMaxF(&amax[c * NHEAD + hh], al);
}

__global__ void k_edge_exp(const int* __restrict__ ei, float* __restrict__ alpha,
                           const float* __restrict__ amax, float* __restrict__ denom) {
    int idx = blockIdx.x * blockDim.x + threadIdx.x;
    if (idx >= ETOT * NHEAD) return;
    int e = idx >> 3, hh = idx & 7;
    int c = (e < NEDGE) ? ei[NEDGE + e] : (e - NEDGE);
    float ex = expf(alpha[idx] - amax[c * NHEAD + hh]);
    alpha[idx] = ex;
    atomicAdd(&denom[c * NHEAD + hh], ex);
}

__global__ void k_bias_init(float* __restrict__ xout, const float* __restrict__ b) {
    int i = blockIdx.x, t = threadIdx.x;
    xout[i * HCDIM + t] = b[t];
}

__global__ void k_aggregate(const int* __restrict__ ei, const float* __restrict__ h,
                            const float* __restrict__ alpha, const float* __restrict__ denom,
                            float* __restrict__ xout) {
    int e = blockIdx.x, t = threadIdx.x;     // t indexes (head,channel) in [0,256)
    int hh = t >> 5;
    int r, c;
    if (e < NEDGE) { r = ei[e]; c = ei[NEDGE + e]; }
    else           { r = c = e - NEDGE; }
    float w = alpha[e * NHEAD + hh] / (denom[c * NHEAD + hh] + 1e-16f);
    atomicAdd(&xout[(size_t)c * HCDIM + t], h[(size_t)r * HCDIM + t] * w);
}

// ---------------------------------------------------------------- WMMA GEMM
// Out[i,o] (+)= sum_k X[i*ldx+k] * W[o*ldw+k].  M,Ncol multiples of 16, K of 4.
// One wave per 16x16 tile; v_wmma_f32_16x16x4_f32; wave-uniform exit keeps EXEC=all-1s.
__global__ void k_gemm_wmma(const float* __restrict__ X, int ldx,
                            const float* __restrict__ W, int ldw,
                            float* __restrict__ Out, int ldo,
                            int M, int Ncol, int K, int accum) {
    const int lane = threadIdx.x & 31;
    const int wave = blockIdx.x * (blockDim.x >> 5) + (threadIdx.x >> 5);
    const int tilesN = Ncol >> 4;
    const int nTiles = (M >> 4) * tilesN;
    if (wave >= nTiles) return;
    const int tm   = (wave / tilesN) << 4;
    const int tn   = (wave % tilesN) << 4;
    const int half = lane >> 4;   // selects K pair {0,1} vs {2,3} (A/B layout §7.12.2)
    const int l15  = lane & 15;

    v8f acc = {0.f, 0.f, 0.f, 0.f, 0.f, 0.f, 0.f, 0.f};
    if (accum) {
#pragma unroll
        for (int v = 0; v < 8; ++v)
            acc[v] = Out[(size_t)(tm + (half << 3) + v) * ldo + tn + l15];
    }
    const float* xr = X + (size_t)(tm + l15) * ldx;
    const float* wr = W + (size_t)(tn + l15) * ldw;
    for (int k = 0; k < K; k += 4) {
        v2f a, b;
        a.x = xr[k + (half << 1)];
        a.y = xr[k + (half << 1) + 1];
        b.x = wr[k + (half << 1)];
        b.y = wr[k + (half << 1) + 1];
        acc = __builtin_amdgcn_wmma_f32_16x16x4_f32(false, a, false, b,
                                                    (short)0, acc, false, false);
    }
#pragma unroll
    for (int v = 0; v < 8; ++v)
        Out[(size_t)(tm + (half << 3) + v) * ldo + tn + l15] = acc[v];
}

// ---------------------------------------------------------------- decoder prep
__global__ void k_AB(const float* __restrict__ x2, const float* __restrict__ Wes,
                     float* __restrict__ A, float* __restrict__ B) {
    int idx = blockIdx.x * blockDim.x + threadIdx.x;   // (i,k)
    if (idx >= N_NODES * 32) return;
    int i = idx >> 5, k = idx & 31;
    const float* xr = x2 + (size_t)i * HCDIM;
    const float* wl = Wes + (size_t)k * (2 * HCDIM);
    const float* wrp = wl + HCDIM;
    float a = 0.0f, b = 0.0f;
    for (int c = 0; c < HCDIM; ++c) { a += xr[c] * wl[c]; b += xr[c] * wrp[c]; }
    A[idx] = a; B[idx] = b;
}

__global__ void k_scatter_stops(const int* __restrict__ stops, float* __restrict__ sv) {
    sv[stops[threadIdx.x]] = 1.0f;
}

__global__ void k_S1(const float* __restrict__ Wc1, float* __restrict__ S1) {
    int idx = blockIdx.x * blockDim.x + threadIdx.x;   // (o,k)
    if (idx >= N_NODES * 32) return;
    int o = idx >> 5, k = idx & 31;
    const float* wr = Wc1 + (size_t)o * IN_LEN + k;
    float s = 0.0f;
    for (int j = 0; j < N_NODES; ++j) s += wr[j * 32];
    S1[idx] = s;
}

__global__ void k_Tconst(const float* __restrict__ Wc1, const float* __restrict__ B,
                         const float* __restrict__ bes, const float* __restrict__ sv,
                         const float* __restrict__ W_week, const float* __restrict__ W_cap,
                         const float* __restrict__ W_veh,
                         const int* __restrict__ weekday, const int* __restrict__ capacity,
                         const int* __restrict__ vehicles,
                         const float* __restrict__ bc1, float* __restrict__ T) {
    __shared__ float sred[256];
    int o = blockIdx.x, t = threadIdx.x;
    const float* wr = Wc1 + (size_t)o * IN_LEN;
    float s = 0.0f;
    for (int jk = t; jk < OFF_DIST; jk += 256)
        s += (B[jk] + bes[jk & 31]) * wr[jk];
    for (int j = t; j < N_NODES; j += 256)
        s += sv[j] * wr[OFF_STOP + j];
    sred[t] = s; __syncthreads();
    for (int st = 128; st > 0; st >>= 1) {
        if (t < st) sred[t] += sred[t + st];
        __syncthreads();
    }
    if (t == 0) {
        float tot = sred[0];
        int wd = weekday[0], cp = capacity[0], vh = vehicles[0];
        for (int f = 0; f < 3; ++f) {
            tot += W_week[wd * 3 + f] * wr[OFF_WEEK + f];
            tot += W_cap [cp * 3 + f] * wr[OFF_CAP  + f];
            tot += W_veh [vh * 3 + f] * wr[OFF_VEH  + f];
        }
        T[o] = tot + bc1[o];
    }
}

__global__ void k_init_hidden(float* __restrict__ hid, const float* __restrict__ T) {
    int idx = blockIdx.x * blockDim.x + threadIdx.x;
    hid[idx] = T[idx & (N_NODES - 1)];
}

__global__ void k_init_out(float* __restrict__ out, const float* __restrict__ bc2) {
    int idx = blockIdx.x * blockDim.x + threadIdx.x;
    out[idx] = bc2[idx & (N_NODES - 1)];
}

// ---------------------------------------------------------------- host side
struct GatWS {
    const int* eidx; const float* ea; const float* la;
    float *a_src, *a_dst, *amax, *denom, *alpha, *wedot;
};

static void run_gat_attention(const float* hbuf, const float* att_src, const float* att_dst,
                              const float* We, const float* att_edge, const float* bias,
                              float* xout, const GatWS& g, hipStream_t s) {
    k_fill<<<16, 256, 0, s>>>(g.amax,  N_NODES * NHEAD, -INFINITY);
    k_fill<<<16, 256, 0, s>>>(g.denom, N_NODES * NHEAD, 0.0f);
    k_attn_scores<<<16, 256, 0, s>>>(hbuf, att_src, att_dst, g.a_src, g.a_dst);
    k_wedot<<<1, 8, 0, s>>>(We, att_edge, g.wedot);
    int eb = (ETOT * NHEAD + 255) / 256;
    k_edge_alpha<<<eb, 256, 0, s>>>(g.eidx, g.ea, g.la, g.a_src, g.a_dst, g.wedot,
                                    g.alpha, g.amax);
    k_edge_exp<<<eb, 256, 0, s>>>(g.eidx, g.alpha, g.amax, g.denom);
    k_bias_init<<<N_NODES, HCDIM, 0, s>>>(xout, bias);
    k_aggregate<<<ETOT, HCDIM, 0, s>>>(g.eidx, hbuf, g.alpha, g.denom, xout);
    k_relu<<<N_NODES, HCDIM, 0, s>>>(xout, N_NODES * HCDIM);
}

extern "C" void kernel_launch(void* const* d_in, const int* in_sizes, int n_in,
                              void* d_out, int out_size, void* d_ws, size_t ws_size,
                              hipStream_t stream) {
    const float* dist    = (const float*)d_in[0];
    const float* markov  = (const float*)d_in[1];
    const float* demand  = (const float*)d_in[2];
    const float* W_week  = (const float*)d_in[3];
    const float* W_cap   = (const float*)d_in[4];
    const float* W_veh   = (const float*)d_in[5];
    const float* g1_W    = (const float*)d_in[6];
    const float* g1_as   = (const float*)d_in[7];
    const float* g1_ad   = (const float*)d_in[8];
    const float* g1_We   = (const float*)d_in[9];
    const float* g1_ae   = (const float*)d_in[10];
    const float* g1_b    = (const float*)d_in[11];
    const float* g2_W    = (const float*)d_in[12];
    const float* g2_as   = (const float*)d_in[13];
    const float* g2_ad   = (const float*)d_in[14];
    const float* g2_We   = (const float*)d_in[15];
    const float* g2_ae   = (const float*)d_in[16];
    const float* g2_b    = (const float*)d_in[17];
    const float* Wes     = (const float*)d_in[18];
    const float* bes     = (const float*)d_in[19];
    const float* Wc1     = (const float*)d_in[20];
    const float* bc1     = (const float*)d_in[21];
    const float* Wc2     = (const float*)d_in[22];
    const float* bc2     = (const float*)d_in[23];
    const int*   stops   = (const int*)d_in[24];
    const int*   eidx    = (const int*)d_in[25];
    const int*   weekday = (const int*)d_in[26];
    const int*   vehicles= (const int*)d_in[27];
    const int*   capacity= (const int*)d_in[28];
    float* out = (float*)d_out;

    // workspace carve-up (floats)
    float* w = (float*)d_ws;
    unsigned* mnmx = (unsigned*)w;            w += 2;
    float* dist_n  = w;                       w += N_NODES * N_NODES;
    float* ea      = w;                       w += NEDGE;
    float* deg     = w;                       w += N_NODES;
    float* loopsum = w;                       w += N_NODES;
    float* la      = w;                       w += N_NODES;
    float* wedot   = w;                       w += NHEAD;
    float* hbuf    = w;                       w += N_NODES * HCDIM;
    float* x1      = w;                       w += N_NODES * HCDIM;
    float* x2      = w;                       w += N_NODES * HCDIM;
    float* a_src   = w;                       w += N_NODES * NHEAD;
    float* a_dst   = w;                       w += N_NODES * NHEAD;
    float* amax    = w;                       w += N_NODES * NHEAD;
    float* denom   = w;                       w += N_NODES * NHEAD;
    float* alpha   = w;                       w += ETOT * NHEAD;
    float* Amat    = w;                       w += N_NODES * 32;
    float* Bmat    = w;                       w += N_NODES * 32;
    float* S1      = w;                       w += N_NODES * 32;
    float* Tconst  = w;                       w += N_NODES;
    float* stopvec = w;                       w += N_NODES;
    float* hidden  = w;                       w += N_NODES * N_NODES;

    // 1) dist normalization
    k_init_minmax<<<1, 1, 0, stream>>>(mnmx);
    k_minmax<<<256, 256, 0, stream>>>(dist, N_NODES * N_NODES, mnmx);
    k_dist_norm<<<1024, 256, 0, stream>>>(dist, mnmx, dist_n);

    // 2) shared edge prep (ea, degree, self-loop attr)
    k_fill<<<2, 256, 0, stream>>>(deg, N_NODES, 0.0f);
    k_fill<<<2, 256, 0, stream>>>(loopsum, N_NODES, 0.0f);
    k_edge_prep<<<NEDGE / 256, 256, 0, stream>>>(eidx, markov, ea, deg, loopsum);
    k_loop_attr<<<2, 256, 0, stream>>>(loopsum, deg, la);

    GatWS g{eidx, ea, la, a_src, a_dst, amax, denom, alpha, wedot};

    // 3) GAT layer 1 (x is N x 1 -> h via outer product)
    k_h_layer1<<<N_NODES, HCDIM, 0, stream>>>(demand, g1_W, hbuf);
    run_gat_attention(hbuf, g1_as, g1_ad, g1_We, g1_ae, g1_b, x1, g, stream);

    // 4) GAT layer 2: h = x1 @ g2_W.T  (WMMA, 512x256x256)
    k_gemm_wmma<<<64, 256, 0, stream>>>(x1, HCDIM, g2_W, HCDIM, hbuf, HCDIM,
                                        N_NODES, HCDIM, HCDIM, 0);
    run_gat_attention(hbuf, g2_as, g2_ad, g2_We, g2_ae, g2_b, x2, g, stream);

    // 5) low-rank decomposition of the er block
    k_AB<<<64, 256, 0, stream>>>(x2, Wes, Amat, Bmat);
    k_fill<<<2, 256, 0, stream>>>(stopvec, N_NODES, 0.0f);
    k_scatter_stops<<<1, 256, 0, stream>>>(stops, stopvec);
    k_S1<<<64, 256, 0, stream>>>(Wc1, S1);
    k_Tconst<<<N_NODES, 256, 0, stream>>>(Wc1, Bmat, bes, stopvec, W_week, W_cap, W_veh,
                                          weekday, capacity, vehicles, bc1, Tconst);

    // 6) hidden = relu( A@S1.T + dist_n@Wd.T + markov@Wm.T + Tconst ) — all WMMA
    k_init_hidden<<<1024, 256, 0, stream>>>(hidden, Tconst);
    k_gemm_wmma<<<128, 256, 0, stream>>>(Amat, 32, S1, 32, hidden, N_NODES,
                                         N_NODES, N_NODES, 32, 1);
    k_gemm_wmma<<<128, 256, 0, stream>>>(dist_n, N_NODES, Wc1 + OFF_DIST, IN_LEN,
                                         hidden, N_NODES, N_NODES, N_NODES, N_NODES, 1);
    k_gemm_wmma<<<128, 256, 0, stream>>>(markov, N_NODES, Wc1 + OFF_MARKOV, IN_LEN,
                                         hidden, N_NODES, N_NODES, N_NODES, N_NODES, 1);
    k_relu<<<1024, 256, 0, stream>>>(hidden, N_NODES * N_NODES);

    // 7) out = hidden @ Wc2.T + bc2 — WMMA
    k_init_out<<<1024, 256, 0, stream>>>(out, bc2);
    k_gemm_wmma<<<128, 256, 0, stream>>>(hidden, N_NODES, Wc2, N_NODES, out, N_NODES,
                                         N_NODES, N_NODES, N_NODES, 1);
}